// STDP_84791244358141
// MI455X (gfx1250) — compile-verified
//
#include <hip/hip_runtime.h>

typedef __attribute__((ext_vector_type(2))) float v2f;
typedef __attribute__((ext_vector_type(8))) float v8f;

// ---------------------------------------------------------------------------
// Kernel 1: leaky-trace scan.
//   x_t = x_{t-1} + (AP*p_t - x_{t-1})/TP   (AP=1, TP=2 -> * 0.5f, exact)
//   y_t likewise with post spikes.
// One thread per column; loads are fully coalesced (thread i reads
// pre[t*N + i]); the 256-step FMA chain is trivially latency-hidden since the
// loads are independent of the recurrence. Results -> ws[0..N) = x_T,
// ws[N..2N) = y_T.
// ---------------------------------------------------------------------------
__global__ void stdp_trace_scan(const float* __restrict__ pre,
                                const float* __restrict__ post,
                                float* __restrict__ ws,
                                int T, int N) {
    int tid = blockIdx.x * blockDim.x + threadIdx.x;
    if (tid < N) {
        const float* col = pre + tid;
        float x = 0.0f;
        #pragma unroll 8
        for (int t = 0; t < T; ++t) {
            x = x + (col[(size_t)t * N] - x) * 0.5f;   // AP=1, 1/TP = 0.5
        }
        ws[tid] = x;
    } else if (tid < 2 * N) {
        int j = tid - N;
        const float* col = post + j;
        float y = 0.0f;
        #pragma unroll 8
        for (int t = 0; t < T; ++t) {
            y = y + (col[(size_t)t * N] - y) * 0.5f;   // AN=1, 1/TN = 0.5
        }
        ws[N + j] = y;
    }
}

// ---------------------------------------------------------------------------
// Kernel 2: E[i,j] = x_T[i]*q_T[j] - p_T[i]*y_T[j]
// Rank-2 GEMM padded to K=4, executed with V_WMMA_F32_16X16X4_F32 so the
// arithmetic stays in exact fp32 (spikes are {0,1} -> products exact).
//
// ISA fragment layouts (CDNA5 ISA, 32-bit A 16x4 / C-D 16x16):
//   A: lane<16 : VGPR0 = A[M=lane][K=0], VGPR1 = A[M=lane][K=1]
//      lane>=16: K=2,3 rows  -> zero padding
//   B: lane<16 : VGPR0 = B[K=0][N=lane], VGPR1 = B[K=1][N=lane]
//      lane>=16: K=2,3 rows  -> zero padding
//   D: VGPR r, lanes 0-15 -> M=r, lanes 16-31 -> M=r+8, N=lane&15
//
// One wave per 16x16 output tile: 64x64 tiles = 4096 waves -> store-BW bound.
// ---------------------------------------------------------------------------
__global__ void stdp_outer_wmma(const float* __restrict__ xw,     // x_T [N]
                                const float* __restrict__ yw,     // y_T [N]
                                const float* __restrict__ pLast,  // p_T [N]
                                const float* __restrict__ qLast,  // q_T [N]
                                float* __restrict__ out,          // [N, N]
                                int N) {
    const int wave   = (blockIdx.x * blockDim.x + threadIdx.x) >> 5;
    const int lane   = threadIdx.x & 31;
    const int tilesN = N >> 4;                 // 64
    const int tm     = wave / tilesN;
    const int tn     = wave - tm * tilesN;
    const int m0     = tm << 4;
    const int n0     = tn << 4;

    const bool lo  = (lane < 16);
    const int  l15 = lane & 15;

    // A fragment: rows m0..m0+15, K = {x_T, -p_T, 0, 0}
    v2f a;
    a.x = lo ?  xw[m0 + l15]    : 0.0f;
    a.y = lo ? -pLast[m0 + l15] : 0.0f;

    // B fragment: cols n0..n0+15, K = {q_T, y_T, 0, 0}
    v2f b;
    b.x = lo ? qLast[n0 + l15] : 0.0f;
    b.y = lo ? yw[n0 + l15]    : 0.0f;

    v8f c = {};
    // (neg_a, A, neg_b, B, c_mod, C, reuse_a, reuse_b) — f32 WMMA has no A/B
    // NEG modifiers (only CNeg), so the subtraction is folded into a.y above.
    c = __builtin_amdgcn_wmma_f32_16x16x4_f32(
            false, a, false, b, (short)0, c, false, false);

    // Scatter the 16x16 tile: VGPR r -> row (m0 + r + 8*(lane>=16)).
    float* o = out + (size_t)(m0 + ((lane >> 4) << 3)) * N + n0 + l15;
    #pragma unroll
    for (int r = 0; r < 8; ++r) {
        o[(size_t)r * N] = c[r];
    }
}

// ---------------------------------------------------------------------------
extern "C" void kernel_launch(void* const* d_in, const int* in_sizes, int n_in,
                              void* d_out, int out_size, void* d_ws, size_t ws_size,
                              hipStream_t stream) {
    const float* pre  = (const float*)d_in[0];   // [T, N]
    const float* post = (const float*)d_in[1];   // [T, N]
    float* out = (float*)d_out;                  // [N, N]
    float* ws  = (float*)d_ws;                   // x_T [N] ++ y_T [N]

    const int N = 1024;
    const int T = in_sizes[0] / N;               // 256

    // Kernel 1: compute final traces x_T, y_T (2N threads).
    {
        const int threads = 256;
        const int blocks  = (2 * N + threads - 1) / threads;  // 8
        stdp_trace_scan<<<blocks, threads, 0, stream>>>(pre, post, ws, T, N);
    }

    // Kernel 2: rank-2 outer product via fp32 WMMA. One wave per 16x16 tile.
    {
        const float* pLast = pre  + (size_t)(T - 1) * N;
        const float* qLast = post + (size_t)(T - 1) * N;
        const int tiles   = (N / 16) * (N / 16);   // 4096 waves
        const int threads = 256;                   // 8 waves/block
        const int blocks  = tiles / (threads / 32); // 512
        stdp_outer_wmma<<<blocks, threads, 0, stream>>>(
            ws, ws + N, pLast, qLast, out, N);
    }
}